// FragNetLayer_22771916603967
// MI455X (gfx1250) — compile-verified
//
#include <hip/hip_runtime.h>
#include <hip/hip_bf16.h>

typedef __attribute__((ext_vector_type(2))) float v2f;
typedef __attribute__((ext_vector_type(8))) float v8f;

#define D 128

// ---------------------------------------------------------------------------
// Small utility kernels
// ---------------------------------------------------------------------------
__global__ void k_zero_f32(float* __restrict__ p, long long n) {
  long long i = (long long)blockIdx.x * blockDim.x + threadIdx.x;
  if (i < n) p[i] = 0.0f;
}

__global__ void k_deg_init(float* __restrict__ deg, int n) {
  int i = blockIdx.x * blockDim.x + threadIdx.x;
  if (i < n) deg[i] = 1.0f;   // self loop contributes 1 to every source degree
}

__global__ void k_deg_accum(const int* __restrict__ src, float* __restrict__ deg, int E) {
  int e = blockIdx.x * blockDim.x + threadIdx.x;
  if (e < E) atomicAdd(&deg[src[e]], 1.0f);
}

__global__ void k_dinv(const float* __restrict__ deg, float* __restrict__ dinv, int n) {
  int i = blockIdx.x * blockDim.x + threadIdx.x;
  if (i < n) {
    float d = deg[i];
    dinv[i] = d > 0.0f ? rsqrtf(d) : 0.0f;
  }
}

// ---------------------------------------------------------------------------
// WMMA f32 GEMM: H[M x 128] = X[M x 128] @ W[128 x 128] + bias
// One block = 16 output rows; 8 waves each own one 16x16 column tile.
// Requires M % 16 == 0 (100000 = 6250*16 here).
//
// V_WMMA_F32_16X16X4_F32 fragment layouts (wave32):
//   A (16x4):  lanes 0-15 M=lane K={0,1}; lanes 16-31 M=lane-16 K={2,3} -> v2f
//   B (4x16):  lanes 0-15 N=lane K={0,1}; lanes 16-31 N=lane-16 K={2,3} -> v2f
//   C/D:       VGPR r: lanes 0-15 -> (M=r, N=lane); lanes 16-31 -> (M=r+8)
// ---------------------------------------------------------------------------
__global__ void k_gemm_h(const float* __restrict__ X, const float* __restrict__ W,
                         const float* __restrict__ bias, float* __restrict__ H) {
  const int row0  = blockIdx.x * 16;
  const int wave  = threadIdx.x >> 5;
  const int lane  = threadIdx.x & 31;
  const int mrow  = lane & 15;
  const int khalf = lane >> 4;          // 0 or 1
  const int n0    = wave * 16;

  v8f acc = {};
#pragma unroll
  for (int k0 = 0; k0 < D; k0 += 4) {
    v2f a = *(const v2f*)(X + (size_t)(row0 + mrow) * D + k0 + 2 * khalf);
    const float* bp = W + (size_t)(k0 + 2 * khalf) * D + n0 + mrow;
    v2f b;
    b.x = bp[0];
    b.y = bp[D];
    acc = __builtin_amdgcn_wmma_f32_16x16x4_f32(false, a, false, b, (short)0, acc,
                                                false, false);
  }
  const int   col = n0 + mrow;
  const float bv  = bias[col];
  const int   rb  = row0 + 8 * khalf;
#pragma unroll
  for (int r = 0; r < 8; ++r)
    H[(size_t)(rb + r) * D + col] = acc[r] + bv;
}

// ---------------------------------------------------------------------------
// Self-loop term initializes the output: out[i][:] = h[i][:] * dinv[i]^2
// ---------------------------------------------------------------------------
__global__ void k_selfloop(const float* __restrict__ H, const float* __restrict__ dinv,
                           float* __restrict__ out, long long total) {
  long long i = (long long)blockIdx.x * blockDim.x + threadIdx.x;
  if (i < total) {
    float di = dinv[i >> 7];
    out[i] = H[i] * di * di;
  }
}

// ---------------------------------------------------------------------------
// Edge scatter: out[tgt] += h[src] * dinv[src]*dinv[tgt]
// One wave per edge; each lane gathers a float4 and issues 4 f32 atomics.
// 51.2 MB destination lives in the 192 MB L2 -> atomics resolve on-chip.
// ---------------------------------------------------------------------------
__global__ void k_edge_scatter(const float* __restrict__ H, const int* __restrict__ src,
                               const int* __restrict__ tgt, const float* __restrict__ dinv,
                               float* __restrict__ out, int E) {
  int e = blockIdx.x * 8 + (threadIdx.x >> 5);
  if (e >= E) return;
  int lane = threadIdx.x & 31;
  int s = src[e], t = tgt[e];
  float norm = dinv[s] * dinv[t];
  float4 v = ((const float4*)(H + (size_t)s * D))[lane];
  float* op = out + (size_t)t * D + lane * 4;
  atomicAdd(op + 0, v.x * norm);
  atomicAdd(op + 1, v.y * norm);
  atomicAdd(op + 2, v.z * norm);
  atomicAdd(op + 3, v.w * norm);
}

// ---------------------------------------------------------------------------
// Pool atoms into fragments: frag_feats[a2f[a]] += x_atoms_new[a]
// ---------------------------------------------------------------------------
__global__ void k_pool_atoms(const float* __restrict__ xnew, const int* __restrict__ a2f,
                             float* __restrict__ frag_feats, int N) {
  int a = blockIdx.x * 8 + (threadIdx.x >> 5);
  if (a >= N) return;
  int lane = threadIdx.x & 31;
  int f = a2f[a];
  float4 v = ((const float4*)(xnew + (size_t)a * D))[lane];
  float* op = frag_feats + (size_t)f * D + lane * 4;
  atomicAdd(op + 0, v.x);
  atomicAdd(op + 1, v.y);
  atomicAdd(op + 2, v.z);
  atomicAdd(op + 3, v.w);
}

// ---------------------------------------------------------------------------
// Fragment edge scatter: frag_sum[ftgt] += frag_feats[fsrc]
// ---------------------------------------------------------------------------
__global__ void k_frag_edge(const float* __restrict__ ff, const int* __restrict__ fsrc,
                            const int* __restrict__ ftgt, float* __restrict__ fsum, int FE) {
  int e = blockIdx.x * 8 + (threadIdx.x >> 5);
  if (e >= FE) return;
  int lane = threadIdx.x & 31;
  int s = fsrc[e], t = ftgt[e];
  float4 v = ((const float4*)(ff + (size_t)s * D))[lane];
  float* op = fsum + (size_t)t * D + lane * 4;
  atomicAdd(op + 0, v.x);
  atomicAdd(op + 1, v.y);
  atomicAdd(op + 2, v.z);
  atomicAdd(op + 3, v.w);
}

// ---------------------------------------------------------------------------
// Fused frag MLP: out = relu(A @ W1 + b1) @ W2 + b2
// A: [M x 128], W1: [128 x 256], W2: [256 x 128]. One block = 16 rows.
// Stage 1: 8 waves x 2 tiles cover the 256-wide hidden; result (bias+ReLU)
// staged in LDS. Stage 2: 8 waves x 1 tile, K=256 from LDS.
// Requires M % 16 == 0 (10000 = 625*16 here).
// ---------------------------------------------------------------------------
__global__ void k_frag_mlp(const float* __restrict__ A, const float* __restrict__ W1,
                           const float* __restrict__ B1, const float* __restrict__ W2,
                           const float* __restrict__ B2, float* __restrict__ out) {
  __shared__ float hid[16 * 256];   // 16 KB of the 320 KB/WGP LDS

  const int row0  = blockIdx.x * 16;
  const int wave  = threadIdx.x >> 5;
  const int lane  = threadIdx.x & 31;
  const int mrow  = lane & 15;
  const int khalf = lane >> 4;

  // ---- Stage 1: hidden = relu(A @ W1 + b1) ----
#pragma unroll
  for (int t = 0; t < 2; ++t) {
    const int n0 = (wave * 2 + t) * 16;
    v8f acc = {};
#pragma unroll
    for (int k0 = 0; k0 < D; k0 += 4) {
      v2f a = *(const v2f*)(A + (size_t)(row0 + mrow) * D + k0 + 2 * khalf);
      const float* bp = W1 + (size_t)(k0 + 2 * khalf) * 256 + n0 + mrow;
      v2f b;
      b.x = bp[0];
      b.y = bp[256];
      acc = __builtin_amdgcn_wmma_f32_16x16x4_f32(false, a, false, b, (short)0, acc,
                                                  false, false);
    }
    const float bv = B1[n0 + mrow];
#pragma unroll
    for (int r = 0; r < 8; ++r) {
      float v = acc[r] + bv;
      hid[(r + 8 * khalf) * 256 + n0 + mrow] = v > 0.0f ? v : 0.0f;
    }
  }
  __syncthreads();

  // ---- Stage 2: out = hidden @ W2 + b2 ----
  {
    const int n0 = wave * 16;
    v8f acc = {};
#pragma unroll
    for (int k0 = 0; k0 < 256; k0 += 4) {
      v2f a = *(const v2f*)&hid[mrow * 256 + k0 + 2 * khalf];
      const float* bp = W2 + (size_t)(k0 + 2 * khalf) * D + n0 + mrow;
      v2f b;
      b.x = bp[0];
      b.y = bp[D];
      acc = __builtin_amdgcn_wmma_f32_16x16x4_f32(false, a, false, b, (short)0, acc,
                                                  false, false);
    }
    const int   col = n0 + mrow;
    const float bv  = B2[col];
    const int   rb  = row0 + 8 * khalf;
#pragma unroll
    for (int r = 0; r < 8; ++r)
      out[(size_t)(rb + r) * D + col] = acc[r] + bv;
  }
}

// ---------------------------------------------------------------------------
// Host-side launch
// ---------------------------------------------------------------------------
extern "C" void kernel_launch(void* const* d_in, const int* in_sizes, int n_in,
                              void* d_out, int out_size, void* d_ws, size_t ws_size,
                              hipStream_t stream) {
  const float* x_atoms  = (const float*)d_in[0];
  const int*   edge_idx = (const int*)d_in[1];
  // d_in[2] edge_attr: dead code in the reference (result unused) -> skipped
  const int*   frag_idx = (const int*)d_in[3];
  // d_in[4] x_frags: only its shape matters
  const int*   a2f      = (const int*)d_in[5];
  const float* atom_w   = (const float*)d_in[6];
  const float* atom_b   = (const float*)d_in[7];
  // d_in[8], d_in[9] edge_w / edge_b: dead code -> skipped
  const float* frag_w1  = (const float*)d_in[10];
  const float* frag_b1  = (const float*)d_in[11];
  const float* frag_w2  = (const float*)d_in[12];
  const float* frag_b2  = (const float*)d_in[13];

  const int N  = in_sizes[5];          // atoms
  const int E  = in_sizes[1] / 2;      // edges
  const int FE = in_sizes[3] / 2;      // frag edges
  const int NF = in_sizes[4] / D;      // fragments

  const int* src  = edge_idx;
  const int* tgt  = edge_idx + E;
  const int* fsrc = frag_idx;
  const int* ftgt = frag_idx + FE;

  // Workspace carve-up (needs ~62.3 MB)
  char*  ws  = (char*)d_ws;
  size_t off = 0;
  float* h = (float*)(ws + off);          off += (size_t)N * D * sizeof(float);
  float* deg = (float*)(ws + off);        off += (size_t)N * sizeof(float);
  float* dinv = (float*)(ws + off);       off += (size_t)N * sizeof(float);
  float* frag_feats = (float*)(ws + off); off += (size_t)NF * D * sizeof(float);
  float* frag_sum = (float*)(ws + off);   off += (size_t)NF * D * sizeof(float);

  float* x_atoms_new = (float*)d_out;                       // [N, D]
  float* x_frags_new = (float*)d_out + (size_t)N * D;       // [NF, D]

  const int TB = 256;

  // 1) degree (self loop = 1) and dinv
  k_deg_init<<<(N + TB - 1) / TB, TB, 0, stream>>>(deg, N);
  k_deg_accum<<<(E + TB - 1) / TB, TB, 0, stream>>>(src, deg, E);
  k_dinv<<<(N + TB - 1) / TB, TB, 0, stream>>>(deg, dinv, N);

  // 2) h = x_atoms @ atom_w + atom_b   (WMMA f32)
  k_gemm_h<<<N / 16, TB, 0, stream>>>(x_atoms, atom_w, atom_b, h);

  // 3) self-loop term initializes the atom output, then edge scatter-add
  long long totA = (long long)N * D;
  k_selfloop<<<(unsigned)((totA + TB - 1) / TB), TB, 0, stream>>>(h, dinv, x_atoms_new, totA);
  k_edge_scatter<<<(E + 7) / 8, TB, 0, stream>>>(h, src, tgt, dinv, x_atoms_new, E);

  // 4) pool atoms -> fragments
  long long totF = (long long)NF * D;
  k_zero_f32<<<(unsigned)((totF + TB - 1) / TB), TB, 0, stream>>>(frag_feats, totF);
  k_pool_atoms<<<(N + 7) / 8, TB, 0, stream>>>(x_atoms_new, a2f, frag_feats, N);

  // 5) fragment-level message passing
  k_zero_f32<<<(unsigned)((totF + TB - 1) / TB), TB, 0, stream>>>(frag_sum, totF);
  k_frag_edge<<<(FE + 7) / 8, TB, 0, stream>>>(frag_feats, fsrc, ftgt, frag_sum, FE);

  // 6) fused frag MLP (WMMA f32 x2, LDS-staged hidden)
  k_frag_mlp<<<NF / 16, TB, 0, stream>>>(frag_sum, frag_w1, frag_b1, frag_w2, frag_b2,
                                         x_frags_new);
}